// NCCLoss_21852793602851
// MI455X (gfx1250) — compile-verified
//
#include <hip/hip_runtime.h>
#include <hip/hip_bf16.h>

// ---------------------------------------------------------------------------
// NCC loss, 9^3 box window, volumes (2,1,192,192,192) fp32.
// Separable plan:  K1: z-sum + y-sum (LDS)  ->  5-channel intermediate (283MB)
//                  K2: x-sum via V_WMMA_F32_16X16X4_F32 + NCC + wave partials
//                  K3: deterministic reduction -> scalar
// HBM traffic ~0.9 GB -> ~38us at 23.3 TB/s; L2 (192MB) absorbs much of the
// intermediate re-read. All math fp32 (box sums ~729 magnitude; f16 too lossy).
// ---------------------------------------------------------------------------

#define DIMV 192
#define PLANE (DIMV * DIMV)                 // 36864
#define VOL   ((size_t)DIMV * DIMV * DIMV)  // 7,077,888
#define NTOT  (2.0f * 192.0f * 192.0f * 192.0f)
#define WINR  4                             // radius, win = 9
#define TY    16                            // y outputs per block in K1
#define YIN   (TY + 2 * WINR)               // 24 input y lines
#define ZIN   (DIMV + 2 * WINR)             // 200: z range incl. halo
#define ZP    201                           // LDS row stride (odd -> bank-spread)

typedef float v2f __attribute__((ext_vector_type(2)));
typedef float v8f __attribute__((ext_vector_type(8)));

// ---------------------------------------------------------------------------
// Kernel 1: fixed (b, x, ytile): load I,J tile [YIN][ZIN] with zeroed y/z halos,
// 9-tap z-sums of {I,J,I*I,J*J,I*J} into LDS (branch-free), 9-tap y-sums,
// store 5 channels at layout [(c*2+b)][x][y][z].
// ---------------------------------------------------------------------------
__global__ __launch_bounds__(256) void ncc_zy_pass(const float* __restrict__ I,
                                                   const float* __restrict__ J,
                                                   float* __restrict__ zy) {
  extern __shared__ float smem[];
  float* sI  = smem;                 // [YIN][ZP], data at z'=0..199 == z=-4..195
  float* sJ  = smem + YIN * ZP;      // [YIN][ZP]
  float* sZS = smem + 2 * YIN * ZP;  // [5][YIN][ZP], data at z=0..191

  const int bx    = blockIdx.x;
  const int ytile = bx % (DIMV / TY);
  const int x     = (bx / (DIMV / TY)) % DIMV;
  const int b     = bx / ((DIMV / TY) * DIMV);
  const int tid   = threadIdx.x;

  const size_t inBase = ((size_t)(b * DIMV + x)) * PLANE;

  // Stage A: load 24 y-lines x 200 z (halo cells -> 0). One pass zeroes both
  // the y-halo (gy OOB) and z-halo (gz OOB), so later stages are branch-free.
  for (int idx = tid; idx < YIN * ZIN; idx += 256) {
    const int yl = idx / ZIN;
    const int zp = idx - yl * ZIN;           // 0..199
    const int gz = zp - WINR;                // -4..195
    const int gy = ytile * TY - WINR + yl;
    float vi = 0.0f, vj = 0.0f;
    if (gy >= 0 && gy < DIMV && gz >= 0 && gz < DIMV) {
      vi = I[inBase + (size_t)gy * DIMV + gz];
      vj = J[inBase + (size_t)gy * DIMV + gz];
    }
    sI[yl * ZP + zp] = vi;
    sJ[yl * ZP + zp] = vj;
  }
  __syncthreads();

  // Stage B: unconditional 9-tap z-sums (z' = z..z+8 covers gz = z-4..z+4)
  for (int idx = tid; idx < YIN * DIMV; idx += 256) {
    const int yl = idx / DIMV;
    const int z  = idx - yl * DIMV;
    float si = 0.f, sj = 0.f, sii = 0.f, sjj = 0.f, sij = 0.f;
    const int base = yl * ZP + z;
#pragma unroll
    for (int k = 0; k < 9; ++k) {
      const float vi = sI[base + k];
      const float vj = sJ[base + k];
      si  += vi;       sj  += vj;
      sii += vi * vi;  sjj += vj * vj;  sij += vi * vj;
    }
    const int o = yl * ZP + z;
    sZS[0 * YIN * ZP + o] = si;
    sZS[1 * YIN * ZP + o] = sj;
    sZS[2 * YIN * ZP + o] = sii;
    sZS[3 * YIN * ZP + o] = sjj;
    sZS[4 * YIN * ZP + o] = sij;
  }
  __syncthreads();

  // Stage C: 9-tap y-sums; output gy = ytile*TY + yo uses tile lines yo..yo+8
  for (int idx = tid; idx < TY * DIMV; idx += 256) {
    const int yo = idx / DIMV;
    const int z  = idx - yo * DIMV;
    const int gy = ytile * TY + yo;
#pragma unroll
    for (int c = 0; c < 5; ++c) {
      float s = 0.f;
#pragma unroll
      for (int dy = 0; dy < 9; ++dy) s += sZS[c * YIN * ZP + (yo + dy) * ZP + z];
      zy[(size_t)(c * 2 + b) * VOL + (size_t)x * PLANE + (size_t)gy * DIMV + z] = s;
    }
  }
}

// ---------------------------------------------------------------------------
// Kernel 2: x-direction 9-tap sum as banded matmul with WMMA f32 16x16x4.
//   out(16x,16z) = sum_{t=0..5} A_t(16x4) * B_t(4x16)
//   A[m][kg] = 1 iff (m <= kg <= m+8) AND input row x0-4+kg is in range;
//   OOB rows are CLAMPED for the load (legal address) and zeroed via A,
//   so every load is unconditional -> no EXEC-mask branches in the stream.
// f32 A layout (ISA 7.12.2): lane -> M = lane&15; VGPR0: K=(lane<16)?0:2,
// VGPR1: K=(lane<16)?1:3. B (4x16) mirrored: N = lane&15, same K striping.
// One wave per 16x16 (x,z) tile at fixed (b,y); 5 channels accumulated.
// ---------------------------------------------------------------------------
__global__ __launch_bounds__(256) void ncc_xpass_wmma(const float* __restrict__ zy,
                                                      float* __restrict__ partials) {
  const int lane  = threadIdx.x & 31;
  const int half  = lane >> 4;       // 0: lanes 0-15, 1: lanes 16-31
  const int n     = lane & 15;       // N (z) / M (x) index within tile
  const int wave  = (int)(blockIdx.x * (blockDim.x >> 5) + (threadIdx.x >> 5));

  const int zt = wave % 12;
  const int xt = (wave / 12) % 12;
  const int y  = (wave / 144) % DIMV;
  const int b  = wave / (144 * DIMV);

  const int z0 = zt * 16;
  const int x0 = xt * 16;

  // Per-channel, per-lane base pointers (row 0, this lane's z column)
  const float* p0 = zy + (size_t)(0 * 2 + b) * VOL + (size_t)y * DIMV + (z0 + n);
  const float* p1 = zy + (size_t)(1 * 2 + b) * VOL + (size_t)y * DIMV + (z0 + n);
  const float* p2 = zy + (size_t)(2 * 2 + b) * VOL + (size_t)y * DIMV + (z0 + n);
  const float* p3 = zy + (size_t)(3 * 2 + b) * VOL + (size_t)y * DIMV + (z0 + n);
  const float* p4 = zy + (size_t)(4 * 2 + b) * VOL + (size_t)y * DIMV + (z0 + n);

  v8f aI  = {0,0,0,0,0,0,0,0};
  v8f aJ  = {0,0,0,0,0,0,0,0};
  v8f aII = {0,0,0,0,0,0,0,0};
  v8f aJJ = {0,0,0,0,0,0,0,0};
  v8f aIJ = {0,0,0,0,0,0,0,0};

#pragma unroll
  for (int t = 0; t < 6; ++t) {
    v2f A, bI, bJ, bII, bJJ, bIJ;
#pragma unroll
    for (int j = 0; j < 2; ++j) {
      const int kg  = 4 * t + j + (half ? 2 : 0);   // global K in 0..23
      const int row = x0 - WINR + kg;               // input x row (may be OOB)
      const bool ok = (row >= 0) && (row < DIMV);
      // band-of-9 AND row-valid; OOB contribution killed here, not at the load
      A[j] = (kg >= n && kg <= n + 8 && ok) ? 1.0f : 0.0f;
      const int rc  = row < 0 ? 0 : (row > DIMV - 1 ? DIMV - 1 : row); // clamp
      const size_t r = (size_t)rc * PLANE;
      bI[j]  = p0[r];
      bJ[j]  = p1[r];
      bII[j] = p2[r];
      bJJ[j] = p3[r];
      bIJ[j] = p4[r];
    }
    aI  = __builtin_amdgcn_wmma_f32_16x16x4_f32(false, A, false, bI,  (short)0, aI,  false, false);
    aJ  = __builtin_amdgcn_wmma_f32_16x16x4_f32(false, A, false, bJ,  (short)0, aJ,  false, false);
    aII = __builtin_amdgcn_wmma_f32_16x16x4_f32(false, A, false, bII, (short)0, aII, false, false);
    aJJ = __builtin_amdgcn_wmma_f32_16x16x4_f32(false, A, false, bJJ, (short)0, aJJ, false, false);
    aIJ = __builtin_amdgcn_wmma_f32_16x16x4_f32(false, A, false, bIJ, (short)0, aIJ, false, false);
  }

  // NCC on the 8 accumulator rows this lane owns (reference formula, fp32)
  const float wsz = 729.0f;
  float lsum = 0.0f;
#pragma unroll
  for (int r = 0; r < 8; ++r) {
    const float Is  = aI[r],  Js  = aJ[r];
    const float I2s = aII[r], J2s = aJJ[r], IJs = aIJ[r];
    const float Im = Is / wsz, Jm = Js / wsz;
    const float cross = IJs - Jm * Is - Im * Js + Im * Jm * wsz;
    const float Ivar  = I2s - 2.0f * Im * Is + Im * Im * wsz;
    const float Jvar  = J2s - 2.0f * Jm * Js + Jm * Jm * wsz;
    lsum += (cross * cross) / (Ivar * Jvar + 1e-5f);
  }

  // wave32 butterfly reduction
#pragma unroll
  for (int sft = 16; sft >= 1; sft >>= 1) lsum += __shfl_xor(lsum, sft, 32);
  if (lane == 0) partials[wave] = lsum;
}

// ---------------------------------------------------------------------------
// Kernel 3: deterministic reduction of partials -> -mean
// ---------------------------------------------------------------------------
__global__ __launch_bounds__(256) void ncc_reduce(const float* __restrict__ p,
                                                  int np, float* __restrict__ out) {
  __shared__ float s[256];
  float acc = 0.0f;
  for (int i = threadIdx.x; i < np; i += 256) acc += p[i];
  s[threadIdx.x] = acc;
  __syncthreads();
  for (int o = 128; o >= 1; o >>= 1) {
    if ((int)threadIdx.x < o) s[threadIdx.x] += s[threadIdx.x + o];
    __syncthreads();
  }
  if (threadIdx.x == 0) out[0] = -(s[0] / NTOT);
}

// ---------------------------------------------------------------------------
extern "C" void kernel_launch(void* const* d_in, const int* in_sizes, int n_in,
                              void* d_out, int out_size, void* d_ws, size_t ws_size,
                              hipStream_t stream) {
  const float* I = (const float*)d_in[0];
  const float* J = (const float*)d_in[1];

  float* zy       = (float*)d_ws;                         // 5*2*VOL floats = 283 MB
  float* partials = (float*)((char*)d_ws + (size_t)10 * VOL * sizeof(float));

  const int nWaves = 2 * DIMV * 12 * 12;                  // 55,296

  // K1: (b=2) x (x=192) x (ytile=12) blocks; ~135 KB dynamic LDS (<320 KB/WGP)
  const size_t lds1 = (size_t)(2 + 5) * YIN * ZP * sizeof(float);
  ncc_zy_pass<<<2 * DIMV * (DIMV / TY), 256, lds1, stream>>>(I, J, zy);

  // K2: 8 waves per block, one 16x16 tile per wave
  ncc_xpass_wmma<<<nWaves / 8, 256, 0, stream>>>(zy, partials);

  // K3: single block, fixed-order reduction (deterministic across replays)
  ncc_reduce<<<1, 256, 0, stream>>>(partials, nWaves, (float*)d_out);
}